// Margin_57732950393273
// MI455X (gfx1250) — compile-verified
//
#include <hip/hip_runtime.h>
#include <hip/hip_bf16.h>

typedef float v2f __attribute__((ext_vector_type(2)));
typedef float v8f __attribute__((ext_vector_type(8)));

#define NPTS    4096
#define KMAX    4
#define DIMS    128
#define TRIP    100000
#define NCLS    100
#define MARGINC 0.2f
#define EPSC    1e-8f

#define CHUNK        16                       // triplets per wave-iteration
#define CHUNKS_PER_K (TRIP / CHUNK)           // 6250
#define TOTAL_CHUNKS (KMAX * CHUNKS_PER_K)    // 25000
#define BLOCKS       500
#define TPB          256
#define WPB          (TPB / 32)

__device__ __forceinline__ void acc_sq(const float4& a, const float4& b, float& s) {
    float d;
    d = a.x - b.x; s = fmaf(d, d, s);
    d = a.y - b.y; s = fmaf(d, d, s);
    d = a.z - b.z; s = fmaf(d, d, s);
    d = a.w - b.w; s = fmaf(d, d, s);
}

__global__ void __launch_bounds__(TPB)
margin_main(const float* __restrict__ batch,
            const int*   __restrict__ labels,
            const int*   __restrict__ triplets,
            const float* __restrict__ beta,
            float*       __restrict__ partials) {
    __shared__ float smem[TPB];

    const int tid   = threadIdx.x;
    const int lane  = tid & 31;
    const int tl    = lane & 15;     // which of the 16 triplets this lane serves
    const int half  = lane >> 4;     // dim half: lanes 0-15 -> dims 0..63, 16-31 -> 64..127
    const int gwave = blockIdx.x * WPB + (tid >> 5);
    const int nwave = gridDim.x * WPB;

    float tot[KMAX] = {0.f, 0.f, 0.f, 0.f};
    float cnt[KMAX] = {0.f, 0.f, 0.f, 0.f};

    for (int c = gwave; c < TOTAL_CHUNKS; c += nwave) {
        const int k = c / CHUNKS_PER_K;
        const int t = (c - k * CHUNKS_PER_K) * CHUNK + tl;

        const int* trip = triplets + ((size_t)k * TRIP + t) * 3;
        const int ia  = trip[0];
        const int ip  = trip[1];
        const int ing = trip[2];

        const float4* Arow = (const float4*)(batch + ((size_t)ia  * KMAX + k) * DIMS) + half * 16;
        const float4* Prow = (const float4*)(batch + ((size_t)ip  * KMAX + k) * DIMS) + half * 16;
        const float4* Nrow = (const float4*)(batch + ((size_t)ing * KMAX + k) * DIMS) + half * 16;

        // two 32-dim partial sums per lane -> exactly the 4x16 f32 B-operand layout
        float sap0 = 0.f, sap1 = 0.f, san0 = 0.f, san1 = 0.f;
        #pragma unroll
        for (int j = 0; j < 16; ++j) {
            float4 av = Arow[j];
            float4 pv = Prow[j];
            float4 nv = Nrow[j];
            if (j < 8) { acc_sq(av, pv, sap0); acc_sq(av, nv, san0); }
            else       { acc_sq(av, pv, sap1); acc_sq(av, nv, san1); }
        }

        // A = ones(16x4), B = partials(4x16): C[i][j] = sum_k B[k][j] = d^2(triplet j).
        // D element 0 then holds d^2(triplet lane&15) in every lane. Exact f32 math.
        v2f ones = {1.0f, 1.0f};
        v2f bap  = {sap0, sap1};
        v2f ban  = {san0, san1};
        v8f zero = {};
        v8f dap = __builtin_amdgcn_wmma_f32_16x16x4_f32(false, ones, false, bap,
                                                        (short)0, zero, false, false);
        v8f dan = __builtin_amdgcn_wmma_f32_16x16x4_f32(false, ones, false, ban,
                                                        (short)0, zero, false, false);

        const float d_ap = sqrtf(dap[0] + EPSC);
        const float d_an = sqrtf(dan[0] + EPSC);
        const float b    = beta[k * NCLS + labels[ia]];
        const float pos  = fmaxf(d_ap - b + MARGINC, 0.f);
        const float neg  = fmaxf(b - d_an + MARGINC, 0.f);

        if (lane < 16) {   // upper half-wave duplicates the per-triplet result
            tot[k] += pos + neg;
            cnt[k] += (pos > 0.f || neg > 0.f) ? 1.f : 0.f;
        }
    }

    // Deterministic block reduction of 8 slots: tot[0..3], cnt[0..3]
    for (int slot = 0; slot < 8; ++slot) {
        const float v = (slot < 4) ? tot[slot] : cnt[slot - 4];
        __syncthreads();
        smem[tid] = v;
        __syncthreads();
        for (int s = TPB / 2; s > 0; s >>= 1) {
            if (tid < s) smem[tid] += smem[tid + s];
            __syncthreads();
        }
        if (tid == 0) partials[blockIdx.x * 8 + slot] = smem[0];
    }
}

__global__ void __launch_bounds__(TPB)
margin_final(const float* __restrict__ partials, float* __restrict__ out) {
    __shared__ float smem[TPB];
    __shared__ float acc[8];
    const int tid = threadIdx.x;

    for (int slot = 0; slot < 8; ++slot) {
        float s = 0.f;
        for (int b = tid; b < BLOCKS; b += TPB) s += partials[b * 8 + slot];
        __syncthreads();
        smem[tid] = s;
        __syncthreads();
        for (int st = TPB / 2; st > 0; st >>= 1) {
            if (tid < st) smem[tid] += smem[tid + st];
            __syncthreads();
        }
        if (tid == 0) acc[slot] = smem[0];
    }
    __syncthreads();
    if (tid == 0) {
        float loss = 0.f;
        for (int k = 0; k < KMAX; ++k) {
            const float tt = acc[k];
            const float cc = acc[4 + k];
            loss += (cc == 0.f) ? tt : tt / fmaxf(cc, 1.f);
        }
        out[0] = loss * 0.25f;
    }
}

extern "C" void kernel_launch(void* const* d_in, const int* in_sizes, int n_in,
                              void* d_out, int out_size, void* d_ws, size_t ws_size,
                              hipStream_t stream) {
    const float* batch    = (const float*)d_in[0];   // (4096, 4, 128) f32
    const int*   labels   = (const int*)d_in[1];     // (4096,)        int
    const int*   triplets = (const int*)d_in[2];     // (4, 100000, 3) int
    const float* beta     = (const float*)d_in[3];   // (4, 100)       f32
    float*       partials = (float*)d_ws;            // BLOCKS * 8 floats

    margin_main<<<BLOCKS, TPB, 0, stream>>>(batch, labels, triplets, beta, partials);
    margin_final<<<1, TPB, 0, stream>>>(partials, (float*)d_out);
}